// CPCLoss_32856499815073
// MI455X (gfx1250) — compile-verified
//
#include <hip/hip_runtime.h>

// CPC loss (CE + BDC + BEC) for N x 128 fp32 logits, single-scalar output.
// Strategy: one wave32 per row. log_sigmoid(l_i - l_j) = (l_i - m) - log(E_i + E_j),
// E = exp(l - m). Symmetric pair enumeration (wrapped distance d=1..63 plus the 64
// antipodal in-register pairs) halves the dominant v_log_f32 count. All wave sum
// reductions use V_WMMA_F32_16X16X4_F32 (A = per-lane values, B = ones -> row-sums).

#define CDIM 128
#define LOG2E 1.44269504088896340736f
#define LN2   0.69314718055994530942f

typedef float v2f_t __attribute__((ext_vector_type(2)));
typedef float v8f_t __attribute__((ext_vector_type(8)));

// Exact f32 sum of one value per lane across the wave, via WMMA.
// With B == all-ones, D[m][n] = rowsum_m(A) for every n. Per the CDNA5 C/D layout,
// lanes 0-15 hold D[0..7][lane] in their 8 VGPRs and lanes 16-31 hold D[8..15][.].
// Horizontal-add of the 8 components gives the half-matrix total; one xor-16
// shuffle-add gives Sum(all 64 A slots) = Sum(per-lane v) in every lane
// (layout-agnostic for A because summation is permutation-invariant).
__device__ __forceinline__ float wave_sum_wmma(float v) {
    v2f_t a; a[0] = v;    a[1] = 0.0f;
    v2f_t b; b[0] = 1.0f; b[1] = 1.0f;
    v8f_t c = {0.0f, 0.0f, 0.0f, 0.0f, 0.0f, 0.0f, 0.0f, 0.0f};
    v8f_t d = __builtin_amdgcn_wmma_f32_16x16x4_f32(
        /*neg_a=*/false, a, /*neg_b=*/false, b,
        /*c_mod=*/(short)0, c, /*reuse_a=*/false, /*reuse_b=*/false);
    float s = ((d[0] + d[1]) + (d[2] + d[3])) + ((d[4] + d[5]) + (d[6] + d[7]));
    s += __shfl_xor(s, 16, 32);
    return s;
}

__device__ __forceinline__ float wave_max(float v) {
    #pragma unroll
    for (int sh = 16; sh >= 1; sh >>= 1) v = fmaxf(v, __shfl_xor(v, sh, 32));
    return v;
}

__global__ __launch_bounds__(256) void cpc_stage1(const float* __restrict__ in,
                                                  const int* __restrict__ tgt,
                                                  float* __restrict__ partial,
                                                  int nrows) {
    // E duplicated (idx and idx+128) so the pair loop needs no index masking.
    __shared__ float Elds[8][2 * CDIM];
    __shared__ float wsum[8];

    const int wave = threadIdx.x >> 5;
    const int lane = threadIdx.x & 31;
    int row = blockIdx.x * 8 + wave;
    const bool valid = row < nrows;
    if (!valid) row = 0;                      // uniform per wave; loads stay in-bounds
    const float* rp = in + (size_t)row * CDIM;

    // Lane owns logits {lane, lane+32, lane+64, lane+96}: coalesced loads,
    // stride-1 LDS bank pattern per instruction.
    const float l0 = rp[lane];
    const float l1 = rp[lane + 32];
    const float l2 = rp[lane + 64];
    const float l3 = rp[lane + 96];

    const float m = wave_max(fmaxf(fmaxf(l0, l1), fmaxf(l2, l3)));

    const float e0 = __expf(l0 - m);
    const float e1 = __expf(l1 - m);
    const float e2 = __expf(l2 - m);
    const float e3 = __expf(l3 - m);

    float* Er = Elds[wave];
    Er[lane]            = e0;  Er[lane + 128] = e0;
    Er[lane + 32]       = e1;  Er[lane + 160] = e1;
    Er[lane + 64]       = e2;  Er[lane + 192] = e2;
    Er[lane + 96]       = e3;  Er[lane + 224] = e3;
    __syncthreads();

    const int   y  = tgt[row];                // uniform scalar load
    const float ly = rp[y];                   // broadcast load
    const float Ey = __expf(ly - m);

    // Per-lane partials (log2 units for everything that goes through v_log_f32).
    float sEp = (e0 + e1) + (e2 + e3);                        // -> Sum E
    float sAp = ((l0 - m) + (l1 - m)) + ((l2 - m) + (l3 - m)); // -> A = Sum(l_i - m)
    float Lyp = (__log2f(Ey + e0) + __log2f(Ey + e1)) +
                (__log2f(Ey + e2) + __log2f(Ey + e3));        // -> L_y (log2 units)

    // d = 64 antipodal pairs {i, i+64} live inside this lane's registers.
    float t0 = __log2f(e0 + e2);
    float t1 = __log2f(e1 + e3);
    float t2 = 0.0f;
    float t3 = 0.0f;

    // Wrapped-distance enumeration: each unordered pair with cyclic distance
    // 1..63 appears exactly once. Duplicated E removes the (idx & 127).
    for (int d = 1; d < 64; ++d) {
        const float a0 = Er[lane + d];
        const float a1 = Er[lane + 32 + d];
        const float a2 = Er[lane + 64 + d];
        const float a3 = Er[lane + 96 + d];
        t0 += __log2f(e0 + a0);
        t1 += __log2f(e1 + a1);
        t2 += __log2f(e2 + a2);
        t3 += __log2f(e3 + a3);
    }
    const float Tp = (t0 + t1) + (t2 + t3);

    // Wave reductions via WMMA (EXEC is all-ones here: no divergence above).
    const float sE = wave_sum_wmma(sEp);
    const float A  = wave_sum_wmma(sAp);
    const float Ly = LN2 * wave_sum_wmma(Lyp);  // back to ln units
    const float T  = LN2 * wave_sum_wmma(Tp);   // Sum_{i<j} log(E_i + E_j)

    // Row algebra (diagonal handled analytically, D = -log2 per row):
    //   ce  = m + log(sE) - l_y
    //   bdc = Sum_{j!=y} log_sig(l_y - l_j)        = 128*dy - Ly + log2
    //   bec = Sum_{i!=y, j!=y} log_sig(l_i - l_j)  = 126*A - 2T - 129*log2 - 128*dy + 2*Ly
    const float dy  = ly - m;
    const float ce  = m + __logf(sE) - ly;
    const float bdc = 128.0f * dy - Ly + LN2;
    const float bec = 126.0f * A - 2.0f * T - 129.0f * LN2 - 128.0f * dy + 2.0f * Ly;
    float contrib = ce - bdc * (1.0f / 127.0f) - bec * (0.5f / 16002.0f);
    if (!valid) contrib = 0.0f;

    if (lane == 0) wsum[wave] = contrib;
    __syncthreads();
    if (threadIdx.x == 0) {
        float t = 0.0f;
        #pragma unroll
        for (int w = 0; w < 8; ++w) t += wsum[w];
        partial[blockIdx.x] = t;
    }
}

__global__ __launch_bounds__(256) void cpc_stage2(const float* __restrict__ partial,
                                                  int n, float* __restrict__ out,
                                                  int nrows) {
    float acc = 0.0f;
    for (int i = threadIdx.x; i < n; i += 256) acc += partial[i];
    const float ws = wave_sum_wmma(acc);      // converged: EXEC all-ones
    __shared__ float tmp[8];
    const int wave = threadIdx.x >> 5;
    const int lane = threadIdx.x & 31;
    if (lane == 0) tmp[wave] = ws;
    __syncthreads();
    if (threadIdx.x == 0) {
        float t = 0.0f;
        #pragma unroll
        for (int w = 0; w < 8; ++w) t += tmp[w];
        out[0] = t / (float)nrows;
    }
}

extern "C" void kernel_launch(void* const* d_in, const int* in_sizes, int n_in,
                              void* d_out, int out_size, void* d_ws, size_t ws_size,
                              hipStream_t stream) {
    const float* in  = (const float*)d_in[0];
    const int*   tgt = (const int*)d_in[1];
    float*       out = (float*)d_out;
    float*       partials = (float*)d_ws;

    const int nrows   = in_sizes[0] / CDIM;   // N = 16384
    const int nblocks = (nrows + 7) / 8;      // 8 rows (waves) per 256-thread block

    cpc_stage1<<<nblocks, 256, 0, stream>>>(in, tgt, partials, nrows);
    cpc_stage2<<<1, 256, 0, stream>>>(partials, nblocks, out, nrows);
}